// Sparsemax_39264591020105
// MI455X (gfx1250) — compile-verified
//
#include <hip/hip_runtime.h>
#include <float.h>

#ifndef __has_builtin
#define __has_builtin(x) 0
#endif

#if __has_builtin(__builtin_amdgcn_global_load_async_to_lds_b128) && \
    __has_builtin(__builtin_amdgcn_s_wait_asynccnt)
#define USE_ASYNC_LDS 1
#else
#define USE_ASYNC_LDS 0
#endif

#define BLOCK    1024            // 32 waves (wave32) per workgroup
#define VEC      8               // 8 x float4 = 32 floats per thread
#define ROW      32000
#define ROW4     8000            // float4s per row
#define CAND_CAP 2048            // candidate buffer (expected ~15 used)

typedef __attribute__((ext_vector_type(4))) float f4v;
typedef __attribute__((__vector_size__(4 * sizeof(int)))) int v4i;  // builtin's pointee type

__device__ __forceinline__ float waveSum(float v) {
#pragma unroll
  for (int o = 16; o > 0; o >>= 1) v += __shfl_xor(v, o, 32);
  return v;
}
__device__ __forceinline__ float waveMax(float v) {
#pragma unroll
  for (int o = 16; o > 0; o >>= 1) v = fmaxf(v, __shfl_xor(v, o, 32));
  return v;
}

extern "C" __global__ void __launch_bounds__(BLOCK, 1)
sparsemax_rows(const float* __restrict__ in, float* __restrict__ out) {
  extern __shared__ float s_row[];          // ROW floats (dynamic, 128000 B)
  __shared__ float    s_red[BLOCK / 32];
  __shared__ float    s_bc;
  __shared__ float    s_cand[CAND_CAP];
  __shared__ unsigned s_cnt;
  __shared__ float    s_tau;

  const int tid = threadIdx.x;
  const int wv  = tid >> 5;
  const int ln  = tid & 31;
  const long long row = blockIdx.x;
  const f4v* __restrict__ g4 = (const f4v*)(in + row * (long long)ROW);
  f4v* __restrict__ o4       = (f4v*)(out + row * (long long)ROW);

  if (tid == 0) s_cnt = 0u;

  f4v v[VEC];
  const f4v NEGBIG = {-FLT_MAX, -FLT_MAX, -FLT_MAX, -FLT_MAX};

#if USE_ASYNC_LDS
  {
    // Stage row Global->LDS via async DMA loads (ASYNCcnt path, bypasses
    // VGPRs), then read back this thread's own chunk; no cross-thread
    // dependency, so only s_wait_asynccnt is required before the DS reads.
    typedef __attribute__((address_space(1))) v4i gv4i;
    typedef __attribute__((address_space(3))) v4i lv4i;
    gv4i* gb = (gv4i*)g4;      // C-style cast: generic->AS1, drops const
    lv4i* lb = (lv4i*)s_row;   // generic->AS3
#pragma unroll
    for (int t = 0; t < VEC; ++t) {
      int f4 = tid + t * BLOCK;
      if (f4 < ROW4) {
        __builtin_amdgcn_global_load_async_to_lds_b128(gb + f4, lb + f4, 0, 0);
      }
    }
    __builtin_amdgcn_s_wait_asynccnt(0);
#pragma unroll
    for (int t = 0; t < VEC; ++t) {
      int f4 = tid + t * BLOCK;
      v[t] = (f4 < ROW4) ? ((const f4v*)s_row)[f4] : NEGBIG;
    }
  }
#else
#pragma unroll
  for (int t = 0; t < VEC; ++t) {
    int f4 = tid + t * BLOCK;
    v[t] = (f4 < ROW4) ? g4[f4] : NEGBIG;
  }
#endif

  // ---- 1) row max (block reduction) ----
  float m = -FLT_MAX;
#pragma unroll
  for (int t = 0; t < VEC; ++t)
    m = fmaxf(m, fmaxf(fmaxf(v[t].x, v[t].y), fmaxf(v[t].z, v[t].w)));
  m = waveMax(m);
  if (ln == 0) s_red[wv] = m;
  __syncthreads();
  if (wv == 0) {
    float x = s_red[ln];                       // BLOCK/32 == 32
    x = waveMax(x);
    if (ln == 0) s_bc = x;
  }
  __syncthreads();
  const float M = s_bc;
  __syncthreads();

  // ---- 2) compact candidates: tau >= M-1 always, so only y = x-M > -1 matter
#pragma unroll
  for (int t = 0; t < VEC; ++t) {
    int f4 = tid + t * BLOCK;
    if (f4 < ROW4) {
      float ys[4] = {v[t].x - M, v[t].y - M, v[t].z - M, v[t].w - M};
#pragma unroll
      for (int c = 0; c < 4; ++c) {
        if (ys[c] > -1.0f) {
          unsigned idx = atomicAdd(&s_cnt, 1u);
          if (idx < (unsigned)CAND_CAP) s_cand[idx] = ys[c];
        }
      }
    }
  }
  __syncthreads();
  const unsigned n = s_cnt;

  // ---- 3) exact threshold via Michelot iteration (no sort) ----
  if (n <= (unsigned)CAND_CAP) {
    if (tid < 32) {                            // single wave solves tiny set
      float S = 0.f;
      for (unsigned i = ln; i < n; i += 32) S += s_cand[i];
      S = waveSum(S);
      float tau = (S - 1.f) / (float)n;
      int k = (int)n;
#pragma unroll 1
      for (int it = 0; it < 64; ++it) {
        float S2 = 0.f, c2 = 0.f;
        for (unsigned i = ln; i < n; i += 32) {
          float c = s_cand[i];
          if (c > tau) { S2 += c; c2 += 1.f; }
        }
        S2 = waveSum(S2);
        c2 = waveSum(c2);
        int k2 = (int)c2;
        tau = (S2 - 1.f) / c2;
        bool done = (k2 == k);
        k = k2;
        if (done) break;                       // support stable -> exact tau
      }
      if (ln == 0) s_tau = tau;
    }
  } else {
    // Fallback (degenerate rows): block-wide Michelot over register data.
    float tau = -1.0f;
    int k = -1;
#pragma unroll 1
    for (int it = 0; it < 64; ++it) {
      __syncthreads();
      float S2 = 0.f, c2 = 0.f;
#pragma unroll
      for (int t = 0; t < VEC; ++t) {
        float ys[4] = {v[t].x - M, v[t].y - M, v[t].z - M, v[t].w - M};
#pragma unroll
        for (int c = 0; c < 4; ++c)
          if (ys[c] > tau) { S2 += ys[c]; c2 += 1.f; }
      }
      S2 = waveSum(S2);
      c2 = waveSum(c2);
      if (ln == 0) s_red[wv] = S2;
      __syncthreads();
      if (wv == 0) { float x = waveSum(s_red[ln]); if (ln == 0) s_bc = x; }
      __syncthreads();
      S2 = s_bc;
      __syncthreads();
      if (ln == 0) s_red[wv] = c2;
      __syncthreads();
      if (wv == 0) { float x = waveSum(s_red[ln]); if (ln == 0) s_bc = x; }
      __syncthreads();
      c2 = s_bc;
      int k2 = (int)c2;
      tau = (S2 - 1.f) / c2;
      bool done = (k2 == k);
      k = k2;
      if (done) break;                         // uniform across block
    }
    if (tid == 0) s_tau = tau;
  }
  __syncthreads();
  const float thr = M + s_tau;                 // out = max(0, x - (M + tau))

  // ---- 4) streaming output (nontemporal: written once, never re-read) ----
#pragma unroll
  for (int t = 0; t < VEC; ++t) {
    int f4 = tid + t * BLOCK;
    if (f4 < ROW4) {
      f4v r;
      r.x = fmaxf(0.f, v[t].x - thr);
      r.y = fmaxf(0.f, v[t].y - thr);
      r.z = fmaxf(0.f, v[t].z - thr);
      r.w = fmaxf(0.f, v[t].w - thr);
      __builtin_nontemporal_store(r, &o4[f4]);
    }
  }
}

extern "C" void kernel_launch(void* const* d_in, const int* in_sizes, int n_in,
                              void* d_out, int out_size, void* d_ws, size_t ws_size,
                              hipStream_t stream) {
  (void)n_in; (void)d_ws; (void)ws_size; (void)out_size;
  const float* in = (const float*)d_in[0];
  float* out      = (float*)d_out;
  const int rows  = in_sizes[0] / ROW;         // 2048
  const size_t shmem = (size_t)ROW * sizeof(float);  // 128000 B dynamic LDS
  sparsemax_rows<<<rows, BLOCK, shmem, stream>>>(in, out);
}